// graph_conv_layer_86784109183525
// MI455X (gfx1250) — compile-verified
//
#include <hip/hip_runtime.h>

typedef __attribute__((ext_vector_type(2))) float v2f;
typedef __attribute__((ext_vector_type(8))) float v8f;

#define BATCH  256
#define NN     512
#define FEAT   128   // IN_FEATURES == U == 128
#define ETYPES 4
// LDS layout for h: element (m,u) at (m>>1)*PS + u*2 + (m&1).
// PS mod 64 == 32 -> half0/half1 lanes of a B-operand ds_load_b64 hit disjoint banks.
#define PS     288   // dwords per interleaved row-pair (2*128 data + 32 pad)

#if __has_builtin(__builtin_amdgcn_tanhf)
#define TANH(x) __builtin_amdgcn_tanhf(x)
#else
#define TANH(x) tanhf(x)
#endif

__device__ __forceinline__ v8f wmma_f32(v2f a, v2f b, v8f c) {
  // D = A(16x4) * B(4x16) + C(16x16), fp32 end-to-end
  return __builtin_amdgcn_wmma_f32_16x16x4_f32(false, a, false, b, (short)0, c,
                                               false, false);
}

extern "C" __global__ __launch_bounds__(512, 1)
void graph_conv_fused_kernel(const float* __restrict__ n_t,   // [B,N,F]
                             const float* __restrict__ adj,   // [B,E,N,N]
                             const float* __restrict__ Wadj,  // [E,F,U]
                             const float* __restrict__ badj,  // [E,U]
                             const float* __restrict__ W2,    // [F,U]
                             const float* __restrict__ b2,    // [U]
                             float* __restrict__ out)         // [B,N,U]
{
  extern __shared__ float h_lds[];   // (NN/2) * PS floats = 288 KB

  const int b    = blockIdx.x;
  const int tid  = threadIdx.x;
  const int w    = tid >> 5;         // wave id 0..15
  const int lane = tid & 31;
  const int half = lane >> 4;        // selects K pair (A/B operands) / M+8 (C tiles)
  const int lm   = lane & 15;
  const int row0 = w * 32;           // this wave's first output row

  // A-operand row base pointers for the two 16-row m-tiles (n_tensor rows)
  const float* nrow0 = n_t + ((size_t)b * NN + row0 + lm) * FEAT + half * 2;
  const float* nrow1 = nrow0 + (size_t)16 * FEAT;

  // Persistent accumulators: acc[mtile][ctile] = 16x16 f32 tile (8 VGPRs each)
  v8f acc[2][8];
  const v8f vzero = {};
#pragma unroll
  for (int t = 0; t < 2; ++t)
#pragma unroll
    for (int c = 0; c < 8; ++c) acc[t][c] = vzero;

  for (int e = 0; e < ETYPES; ++e) {
    // ---- Phase 1: h[b,e] = n[b] @ W_adj[e] + b_adj[e]  -> LDS (this wave's 32 rows)
    const float* We = Wadj + (size_t)e * FEAT * FEAT + lm;
#pragma unroll 1
    for (int c = 0; c < 8; ++c) {
      const float bias = badj[e * FEAT + c * 16 + lm];
      v8f h0, h1;
#pragma unroll
      for (int r = 0; r < 8; ++r) { h0[r] = bias; h1[r] = bias; }
      const float* wp = We + c * 16;
#pragma unroll 1
      for (int k = 0; k < FEAT; k += 4) {
        v2f a0 = *(const v2f*)(nrow0 + k);
        v2f a1 = *(const v2f*)(nrow1 + k);
        v2f bb;
        bb.x = wp[(size_t)(k + half * 2 + 0) * FEAT];
        bb.y = wp[(size_t)(k + half * 2 + 1) * FEAT];
        h0 = wmma_f32(a0, bb, h0);
        h1 = wmma_f32(a1, bb, h1);
      }
      // Store to interleaved LDS layout: rows r,r+1 -> consecutive dwords (b64 stores)
      const int scol = (c * 16 + lm) * 2;
      float* sp0 = h_lds + ((row0 + half * 8) >> 1) * PS + scol;
      float* sp1 = sp0 + 8 * PS;   // +16 rows
#pragma unroll
      for (int q = 0; q < 4; ++q) {
        v2f s0, s1;
        s0.x = h0[2 * q]; s0.y = h0[2 * q + 1];
        s1.x = h1[2 * q]; s1.y = h1[2 * q + 1];
        *(v2f*)(sp0 + q * PS) = s0;
        *(v2f*)(sp1 + q * PS) = s1;
      }
    }
    __syncthreads();

    // ---- Phase 2: acc += adj[b,e,rows,:] @ h_lds  (K = 512, step 4)
    const float* arow0 =
        adj + (((size_t)b * ETYPES + e) * NN + row0 + lm) * NN + half * 2;
    const float* arow1 = arow0 + (size_t)16 * NN;
#pragma unroll 1
    for (int m = 0; m < NN; m += 4) {
      v2f a0 = *(const v2f*)(arow0 + m);
      v2f a1 = *(const v2f*)(arow1 + m);
      // B pair {h[K][u], h[K+1][u]}, K = m + 2*half: consecutive dwords -> ds_load_b64
      const float* bp = h_lds + ((m >> 1) + half) * PS + lm * 2;
#pragma unroll
      for (int c = 0; c < 8; ++c) {
        v2f bb = *(const v2f*)(bp + c * 32);
        acc[0][c] = wmma_f32(a0, bb, acc[0][c]);
        acc[1][c] = wmma_f32(a1, bb, acc[1][c]);
      }
    }
    __syncthreads();   // before next e overwrites h_lds
  }

  // ---- Phase 3: acc += n[b] @ W2 + b2 (c fully unrolled: acc stays in VGPRs)
#pragma unroll
  for (int c = 0; c < 8; ++c) {
    const float bias = b2[c * 16 + lm];
#pragma unroll
    for (int t = 0; t < 2; ++t)
#pragma unroll
      for (int r = 0; r < 8; ++r) acc[t][c][r] += bias;
    const float* wp = W2 + c * 16 + lm;
#pragma unroll 1
    for (int k = 0; k < FEAT; k += 4) {
      v2f a0 = *(const v2f*)(nrow0 + k);
      v2f a1 = *(const v2f*)(nrow1 + k);
      v2f bb;
      bb.x = wp[(size_t)(k + half * 2 + 0) * FEAT];
      bb.y = wp[(size_t)(k + half * 2 + 1) * FEAT];
      acc[0][c] = wmma_f32(a0, bb, acc[0][c]);
      acc[1][c] = wmma_f32(a1, bb, acc[1][c]);
    }
  }

  // ---- Phase 4: tanh + store (C layout: reg r -> row = t*16 + half*8 + r, col = c*16+lm)
#pragma unroll
  for (int t = 0; t < 2; ++t)
#pragma unroll
    for (int r = 0; r < 8; ++r) {
      float* op = out + ((size_t)b * NN + row0 + t * 16 + half * 8 + r) * FEAT + lm;
#pragma unroll
      for (int c = 0; c < 8; ++c) op[c * 16] = TANH(acc[t][c][r]);
    }
}

extern "C" void kernel_launch(void* const* d_in, const int* in_sizes, int n_in,
                              void* d_out, int out_size, void* d_ws, size_t ws_size,
                              hipStream_t stream) {
  const float* n_t  = (const float*)d_in[0];
  const float* adj  = (const float*)d_in[1];
  const float* Wadj = (const float*)d_in[2];
  const float* badj = (const float*)d_in[3];
  const float* W2   = (const float*)d_in[4];
  const float* b2   = (const float*)d_in[5];
  float* out = (float*)d_out;
  (void)in_sizes; (void)n_in; (void)out_size; (void)d_ws; (void)ws_size;

  const size_t lds_bytes = (size_t)(NN / 2) * PS * sizeof(float);  // 294912 B < 320 KB
  hipLaunchKernelGGL(graph_conv_fused_kernel, dim3(BATCH), dim3(512), lds_bytes,
                     stream, n_t, adj, Wadj, badj, W2, b2, out);
}